// TemporalMatchingNetwork_64622077936317
// MI455X (gfx1250) — compile-verified
//
#include <hip/hip_runtime.h>
#include <hip/hip_bf16.h>

typedef __attribute__((ext_vector_type(2))) float v2f;
typedef __attribute__((ext_vector_type(8))) float v8f;

#define WMMA_F32(a, b, c) \
    __builtin_amdgcn_wmma_f32_16x16x4_f32(false, (a), false, (b), (short)0, (c), false, false)

// ---------------------------------------------------------------------------
// Kernel 0: temporal mean over T=4 frames.  x:[4][N], out:[N]
// ---------------------------------------------------------------------------
__global__ __launch_bounds__(256) void k_mean(const float* __restrict__ x,
                                              float* __restrict__ out,
                                              int n, int strideT) {
    int i = blockIdx.x * blockDim.x + threadIdx.x;
    if (i < n) {
        out[i] = 0.25f * (x[i] + x[i + strideT] + x[i + 2 * strideT] + x[i + 3 * strideT]);
    }
}

// ---------------------------------------------------------------------------
// Kernel 1: projection GEMM.  A:[M x 1024] row-major, W:[3][1024][256],
// bias:[3][256] -> out:[3][M][256] (raw, pre-normalize).
// One wave (32 threads) per (h, 16-row tile); 16 accumulators = 16x256 tile.
// __launch_bounds__(32,1): full per-wave VGPR budget, no accumulator spills.
// ---------------------------------------------------------------------------
__global__ __launch_bounds__(32, 1) void k_proj(const float* __restrict__ A,
                                                const float* __restrict__ W,
                                                const float* __restrict__ bias,
                                                float* __restrict__ out, int M) {
    const int K = 1024, P = 256;
    int mtiles = M >> 4;
    int wid  = blockIdx.x;          // one wave per block
    int h    = wid / mtiles;
    int mt   = wid % mtiles;
    int lane = threadIdx.x;
    int half = lane >> 4;           // 0: K pair {0,1}; 1: K pair {2,3}
    int l16  = lane & 15;

    v8f c[16] = {};
    const float* Arow = A + (size_t)(mt * 16 + l16) * K;
    const float* Wh   = W + (size_t)h * K * P;

    for (int kt = 0; kt < K / 4; ++kt) {
        int k0 = kt * 4 + half * 2;
        v2f a = *(const v2f*)(Arow + k0);
        const float* Bp = Wh + (size_t)k0 * P + l16;
        v2f b[16];
#pragma unroll
        for (int nt = 0; nt < 16; ++nt) {
            b[nt].x = Bp[nt * 16];
            b[nt].y = Bp[nt * 16 + P];
        }
#pragma unroll
        for (int nt = 0; nt < 16; ++nt) {
            c[nt] = WMMA_F32(a, b[nt], c[nt]);
        }
    }

#pragma unroll
    for (int nt = 0; nt < 16; ++nt) {
        int p  = nt * 16 + l16;
        float bv = bias[h * P + p];
#pragma unroll
        for (int r = 0; r < 8; ++r) {
            int m = mt * 16 + half * 8 + r;
            out[((size_t)h * M + m) * P + p] = c[nt][r] + bv;
        }
    }
}

// ---------------------------------------------------------------------------
// Kernel 2: in-place L2 row normalization, row length 256, one wave per row.
// Torch cosine_similarity eps clamp: x / max(||x||, 1e-8).
// ---------------------------------------------------------------------------
__global__ __launch_bounds__(256) void k_norm(float* __restrict__ x, int rows) {
    int row  = blockIdx.x * (blockDim.x >> 5) + (threadIdx.x >> 5);
    int lane = threadIdx.x & 31;
    if (row >= rows) return;
    float* p = x + (size_t)row * 256;

    float v[8];
    float s = 0.0f;
#pragma unroll
    for (int j = 0; j < 8; ++j) {
        v[j] = p[lane + 32 * j];
        s += v[j] * v[j];
    }
#pragma unroll
    for (int off = 16; off > 0; off >>= 1) s += __shfl_xor(s, off);
    float inv = 1.0f / fmaxf(sqrtf(s), 1e-8f);
#pragma unroll
    for (int j = 0; j < 8; ++j) p[lane + 32 * j] = v[j] * inv;
}

// ---------------------------------------------------------------------------
// Kernel 3: cosine-sim GEMM + mean over heads.
// qn:[3][4096][256], sn:[3][2048][256] (both L2-normalized, row-major over P)
// out:[4096][2048] = (1/3) * sum_h qn[h] * sn[h]^T
// One wave per block computes a 16(q) x 256(s) tile (16 accumulators).
// Inner-product GEMM: both A and B fragments are contiguous float2 loads.
// ---------------------------------------------------------------------------
__global__ __launch_bounds__(32, 1) void k_dist(const float* __restrict__ qn,
                                                const float* __restrict__ sn,
                                                float* __restrict__ out) {
    const int P = 256, Nq = 4096, Ns = 2048, H = 3;
    int wid  = blockIdx.x;          // 0..2047
    int qt   = wid >> 3;            // 0..255  (16 query rows each)
    int sb   = wid & 7;             // 0..7    (256 support cols each)
    int lane = threadIdx.x;
    int half = lane >> 4;
    int l16  = lane & 15;

    v8f c[16] = {};
    for (int h = 0; h < H; ++h) {
        const float* Aq = qn + ((size_t)h * Nq + qt * 16 + l16) * P;
        const float* Bs = sn + ((size_t)h * Ns + sb * 256 + l16) * P;
        for (int kt = 0; kt < P / 4; ++kt) {
            int k0 = kt * 4 + half * 2;
            v2f a = *(const v2f*)(Aq + k0);
            v2f b[16];
#pragma unroll
            for (int nt = 0; nt < 16; ++nt) {
                b[nt] = *(const v2f*)(Bs + (size_t)nt * 16 * P + k0);
            }
#pragma unroll
            for (int nt = 0; nt < 16; ++nt) {
                c[nt] = WMMA_F32(a, b[nt], c[nt]);
            }
        }
    }

    const float scale = 1.0f / 3.0f;
#pragma unroll
    for (int nt = 0; nt < 16; ++nt) {
        int n = sb * 256 + nt * 16 + l16;
#pragma unroll
        for (int r = 0; r < 8; ++r) {
            int m = qt * 16 + half * 8 + r;
            out[(size_t)m * Ns + n] = c[nt][r] * scale;
        }
    }
}

// ---------------------------------------------------------------------------
// Kernel 4: classifier GEMM.  S:[2048][1024] @ Wc:[1024][100] + bc -> [2048][100]
// N=100 -> 7 column tiles, last one masked.
// ---------------------------------------------------------------------------
__global__ __launch_bounds__(32, 1) void k_cls(const float* __restrict__ S,
                                               const float* __restrict__ Wc,
                                               const float* __restrict__ bc,
                                               float* __restrict__ out) {
    const int K = 1024, C = 100;
    int mt   = blockIdx.x;          // 128 tiles
    int lane = threadIdx.x;
    int half = lane >> 4;
    int l16  = lane & 15;

    v8f c[7] = {};
    const float* Arow = S + (size_t)(mt * 16 + l16) * K;

    for (int kt = 0; kt < K / 4; ++kt) {
        int k0 = kt * 4 + half * 2;
        v2f a = *(const v2f*)(Arow + k0);
        v2f b[7];
#pragma unroll
        for (int nt = 0; nt < 7; ++nt) {
            int col = nt * 16 + l16;
            if (col < C) {
                b[nt].x = Wc[(size_t)k0 * C + col];
                b[nt].y = Wc[(size_t)(k0 + 1) * C + col];
            } else {
                b[nt].x = 0.0f;
                b[nt].y = 0.0f;
            }
        }
#pragma unroll
        for (int nt = 0; nt < 7; ++nt) {
            c[nt] = WMMA_F32(a, b[nt], c[nt]);
        }
    }

#pragma unroll
    for (int nt = 0; nt < 7; ++nt) {
        int col = nt * 16 + l16;
        if (col < C) {
            float bv = bc[col];
#pragma unroll
            for (int r = 0; r < 8; ++r) {
                int m = mt * 16 + half * 8 + r;
                out[(size_t)m * C + col] = c[nt][r] + bv;
            }
        }
    }
}

// ---------------------------------------------------------------------------
extern "C" void kernel_launch(void* const* d_in, const int* in_sizes, int n_in,
                              void* d_out, int out_size, void* d_ws, size_t ws_size,
                              hipStream_t stream) {
    const int Nq = 4096, D = 1024, Ns = 2048, H = 3, P = 256, C = 100;

    const float* query = (const float*)d_in[0];   // [T][Nq][D]
    const float* supp  = (const float*)d_in[1];   // [Ns][D]
    const float* headW = (const float*)d_in[2];   // [H][D][P]
    const float* headB = (const float*)d_in[3];   // [H][P]
    const float* clsW  = (const float*)d_in[4];   // [D][C]
    const float* clsB  = (const float*)d_in[5];   // [C]

    float* out_match = (float*)d_out;                         // [Nq][Ns]
    float* out_preds = out_match + (size_t)Nq * Ns;           // [Ns][C]

    // Workspace layout (floats): agg[Nq*D] | qn[H*Nq*P] | sn[H*Ns*P]  (~34 MB)
    float* agg = (float*)d_ws;
    float* qn  = agg + (size_t)Nq * D;
    float* sn  = qn + (size_t)H * Nq * P;

    // 1) temporal mean
    {
        int n = Nq * D;
        k_mean<<<(n + 255) / 256, 256, 0, stream>>>(query, agg, n, n);
    }

    // 2) head projections (raw, +bias)
    k_proj<<<H * (Nq / 16), 32, 0, stream>>>(agg,  headW, headB, qn, Nq);
    k_proj<<<H * (Ns / 16), 32, 0, stream>>>(supp, headW, headB, sn, Ns);

    // 3) L2 normalize (qn and sn are contiguous in ws: one pass over all rows)
    {
        int rows = H * Nq + H * Ns;          // 18432 rows of length 256
        int rows_per_block = 256 / 32;
        k_norm<<<(rows + rows_per_block - 1) / rows_per_block, 256, 0, stream>>>(qn, rows);
    }

    // 4) cosine GEMM + head mean (one wave per 16x256 output tile)
    k_dist<<<(Nq / 16) * (Ns / 256), 32, 0, stream>>>(qn, sn, out_match);

    // 5) classifier
    k_cls<<<Ns / 16, 32, 0, stream>>>(supp, clsW, clsB, out_preds);
}